// GCN_23673859735705
// MI455X (gfx1250) — compile-verified
//
#include <hip/hip_runtime.h>

typedef __attribute__((ext_vector_type(2))) float v2f;
typedef __attribute__((ext_vector_type(8))) float v8f;

#define HCH 128
#define OCH 64

// ---------------- degree / normalization ----------------
__global__ __launch_bounds__(256) void k_fill_deg(float* __restrict__ deg, int n) {
  int i = blockIdx.x * 256 + threadIdx.x;
  if (i < n) deg[i] = 1.0f;  // self-loop contributes 1
}

__global__ __launch_bounds__(256) void k_count_deg(const int* __restrict__ dst,
                                                   float* __restrict__ deg, int e) {
  int i = blockIdx.x * 256 + threadIdx.x;
  if (i < e) atomicAdd(&deg[dst[i]], 1.0f);
}

__global__ __launch_bounds__(256) void k_dinv(float* __restrict__ deg, int n) {
  int i = blockIdx.x * 256 + threadIdx.x;
  if (i < n) deg[i] = rsqrtf(deg[i]);  // deg >= 1 always
}

// ---------------- GEMM1: h = x @ W1  (N x 128 * 128 x 128) ----------------
// One wave computes one 16x16 output tile. Block = 8 waves = 8 column tiles
// of the same 16-row stripe (shared x rows hit L0/L2). grid.x = N/16.
// Fused epilogue: also writes agg = h * dinv(row)^2 (self-loop init).
__global__ __launch_bounds__(256) void k_gemm1(const float* __restrict__ X,
                                               const float* __restrict__ W,
                                               const float* __restrict__ dinv,
                                               float* __restrict__ Hout,
                                               float* __restrict__ Agg, int n) {
  const int wave = threadIdx.x >> 5;
  const int lane = threadIdx.x & 31;
  const int rowTile = blockIdx.x;
  const int colTile = wave;                 // 0..7  (8 * 16 = 128 cols)
  if (rowTile * 16 >= n) return;            // wave-uniform
  const int hs = lane >> 4;                 // half-wave select (K pair / M+8)
  const int lr = lane & 15;
  const int row = rowTile * 16 + lr;        // A-matrix row for this lane
  const int col = colTile * 16 + lr;        // B/C column for this lane
  const float* __restrict__ xr = X + (size_t)row * HCH;

  v8f acc = {};
#pragma unroll 4
  for (int k0 = 0; k0 < HCH; k0 += 4) {
    const int ka = k0 + hs * 2;             // lanes 0-15: K0/K1, lanes 16-31: K2/K3
    v2f a; a.x = xr[ka];             a.y = xr[ka + 1];
    v2f b; b.x = W[ka * HCH + col];  b.y = W[(ka + 1) * HCH + col];
    acc = __builtin_amdgcn_wmma_f32_16x16x4_f32(false, a, false, b,
                                                (short)0, acc, false, false);
  }
#pragma unroll
  for (int r = 0; r < 8; ++r) {
    const int m = rowTile * 16 + hs * 8 + r;  // C/D layout: VGPR r -> M=r / M=r+8
    const float v = acc[r];
    const float di = dinv[m];
    Hout[(size_t)m * HCH + col] = v;
    Agg [(size_t)m * HCH + col] = v * di * di;   // self-loop term
  }
}

// ---------------- GEMM2: h2 = relu(agg + b1) @ W2  (N x 128 * 128 x 64) --------
// Block = 8 waves = 2 row tiles x 4 column tiles. grid.x = N/32.
// Fused prologue: bias + ReLU applied on the A-matrix loads.
__global__ __launch_bounds__(256) void k_gemm2(const float* __restrict__ Agg,
                                               const float* __restrict__ b1,
                                               const float* __restrict__ W,
                                               float* __restrict__ H2, int n) {
  const int wave = threadIdx.x >> 5;
  const int lane = threadIdx.x & 31;
  const int rowTile = blockIdx.x * 2 + (wave >> 2);
  const int colTile = wave & 3;             // 0..3  (4 * 16 = 64 cols)
  if (rowTile * 16 >= n) return;            // wave-uniform
  const int hs = lane >> 4;
  const int lr = lane & 15;
  const int row = rowTile * 16 + lr;
  const int col = colTile * 16 + lr;
  const float* __restrict__ ar = Agg + (size_t)row * HCH;

  v8f acc = {};
#pragma unroll 4
  for (int k0 = 0; k0 < HCH; k0 += 4) {
    const int ka = k0 + hs * 2;
    float a0 = ar[ka]     + b1[ka];
    float a1 = ar[ka + 1] + b1[ka + 1];
    v2f a; a.x = a0 > 0.0f ? a0 : 0.0f;
           a.y = a1 > 0.0f ? a1 : 0.0f;
    v2f b; b.x = W[ka * OCH + col];  b.y = W[(ka + 1) * OCH + col];
    acc = __builtin_amdgcn_wmma_f32_16x16x4_f32(false, a, false, b,
                                                (short)0, acc, false, false);
  }
#pragma unroll
  for (int r = 0; r < 8; ++r) {
    const int m = rowTile * 16 + hs * 8 + r;
    H2[(size_t)m * OCH + col] = acc[r];
  }
}

// ---------------- conv1 edge aggregation (one wave per edge, float4) ----------
__global__ __launch_bounds__(256) void k_agg1_edges(const int* __restrict__ src,
                                                    const int* __restrict__ dst,
                                                    const float* __restrict__ dinv,
                                                    const float* __restrict__ h,
                                                    float* __restrict__ agg, int e) {
  int gid = blockIdx.x * 256 + threadIdx.x;   // E*32 threads: 32 lanes x 4 ch/lane
  int eidx = gid >> 5;
  if (eidx >= e) return;
  int c4 = (gid & 31) * 4;
  int s = src[eidx], d = dst[eidx];
  float nrm = dinv[s] * dinv[d];
  const float4 hv = *(const float4*)(h + (size_t)s * HCH + c4);
  float* ap = agg + (size_t)d * HCH + c4;
  atomicAdd(ap + 0, hv.x * nrm);
  atomicAdd(ap + 1, hv.y * nrm);
  atomicAdd(ap + 2, hv.z * nrm);
  atomicAdd(ap + 3, hv.w * nrm);
}

// ---------------- conv2 aggregation ----------------
__global__ __launch_bounds__(256) void k_out_init(float* __restrict__ out,
                                                  const float* __restrict__ b2, int total) {
  int gid = blockIdx.x * 256 + threadIdx.x;
  if (gid < total) out[gid] = b2[gid & (OCH - 1)];
}

__global__ __launch_bounds__(256) void k_agg2_edges(const int* __restrict__ src,
                                                    const int* __restrict__ dst,
                                                    const float* __restrict__ h2,
                                                    float* __restrict__ out, int e) {
  int gid = blockIdx.x * 256 + threadIdx.x;   // E*16 threads: 16 lanes x 4 ch/lane
  int eidx = gid >> 4;
  if (eidx >= e) return;
  int c4 = (gid & 15) * 4;
  int s = src[eidx], d = dst[eidx];
  const float4 hv = *(const float4*)(h2 + (size_t)s * OCH + c4);
  float* op = out + (size_t)d * OCH + c4;
  atomicAdd(op + 0, hv.x);
  atomicAdd(op + 1, hv.y);
  atomicAdd(op + 2, hv.z);
  atomicAdd(op + 3, hv.w);
}

// ---------------- launch ----------------
extern "C" void kernel_launch(void* const* d_in, const int* in_sizes, int n_in,
                              void* d_out, int out_size, void* d_ws, size_t ws_size,
                              hipStream_t stream) {
  const float* x  = (const float*)d_in[0];
  const int*   ei = (const int*)d_in[1];
  const float* W1 = (const float*)d_in[2];
  const float* b1 = (const float*)d_in[3];
  const float* W2 = (const float*)d_in[4];
  const float* b2 = (const float*)d_in[5];

  const int N = in_sizes[0] / HCH;
  const int E = in_sizes[1] / 2;
  const int* src = ei;
  const int* dst = ei + E;

  // workspace layout (floats): dinv[N] | h[N*128] | agg[N*128]; h2 reuses h
  float* ws   = (float*)d_ws;
  float* dinv = ws;
  size_t off  = ((size_t)N + 63) & ~(size_t)63;
  float* h    = ws + off;
  float* agg  = h + (size_t)N * HCH;
  float* h2   = h;                       // h is dead after conv1 aggregation
  float* out  = (float*)d_out;

  // 1) symmetric normalization coefficients
  k_fill_deg <<<(N + 255) / 256, 256, 0, stream>>>(dinv, N);
  k_count_deg<<<(E + 255) / 256, 256, 0, stream>>>(dst, dinv, E);
  k_dinv     <<<(N + 255) / 256, 256, 0, stream>>>(dinv, N);

  // 2) h = x @ W1 (WMMA f32), fused self-loop init of agg
  k_gemm1<<<N / 16, 256, 0, stream>>>(x, W1, dinv, h, agg, N);

  // 3) conv1 edge scatter with symmetric norm (one wave per edge, b128 gathers)
  long long ew1 = (long long)E * 32;
  k_agg1_edges<<<(unsigned)((ew1 + 255) / 256), 256, 0, stream>>>(src, dst, dinv, h, agg, E);

  // 4) h2 = relu(agg + b1) @ W2 (WMMA f32, fused bias+ReLU on A loads)
  k_gemm2<<<N / 32, 256, 0, stream>>>(agg, b1, W2, h2, N);

  // 5) conv2 aggregation: out = b2 + scatter(h2)
  int tot2 = N * OCH;
  k_out_init  <<<(tot2 + 255) / 256, 256, 0, stream>>>(out, b2, tot2);
  long long ew2 = (long long)E * 16;
  k_agg2_edges<<<(unsigned)((ew2 + 255) / 256), 256, 0, stream>>>(src, dst, h2, out, E);
}